// ChamferDistanceLoss_3444563771754
// MI455X (gfx1250) — compile-verified
//
#include <hip/hip_runtime.h>
#include <math.h>

typedef __attribute__((ext_vector_type(2))) float v2f;
typedef __attribute__((ext_vector_type(8))) float v8f;

#define BATCH 4
#define NPTS 8192                 // N == M == 8192 per reference
#define ROWS_PER_WAVE 16
#define WAVES_PER_BLOCK 8
#define ROWS_PER_BLOCK (ROWS_PER_WAVE * WAVES_PER_BLOCK)  // 128

// ---------------------------------------------------------------------------
// Bare min ops via inline asm: avoids the IEEE sNaN canonicalization
// (v_max_num_f32 x,x,x) the compiler wraps around fminf. Inputs here are
// finite products of finite data, so minimumNumber semantics are exact.
// ---------------------------------------------------------------------------
__device__ __forceinline__ float min3_asm(float acc, float x, float y) {
    asm("v_min3_num_f32 %0, %0, %1, %2" : "+v"(acc) : "v"(x), "v"(y));
    return acc;
}
__device__ __forceinline__ float min2_asm(float acc, float x) {
    asm("v_min_num_f32 %0, %0, %1" : "+v"(acc) : "v"(x));
    return acc;
}

// ds_swizzle group-of-32 XOR patterns: offset = {xor[14:10], or[9:5], and[4:0]}
#define SWZ_XOR1 0x041F
#define SWZ_XOR2 0x081F
#define SWZ_XOR4 0x101F
#define SWZ_XOR8 0x201F

#define SWZ_MIN(v, pat)                                                     \
    do {                                                                    \
        int _iv = __builtin_bit_cast(int, (v));                             \
        int _sw = __builtin_amdgcn_ds_swizzle(_iv, (pat));                  \
        (v) = min2_asm((v), __builtin_bit_cast(float, _sw));                \
    } while (0)

// ---------------------------------------------------------------------------
// Prep: for each point p=(x,y,z) of both clouds, pack the "B-operand" float4
//   b4 = (-2x, -2y, -2z, x^2+y^2+z^2)
// and stash n2 = ||p||^2 separately (added per-row after the min).
// ---------------------------------------------------------------------------
__global__ __launch_bounds__(256) void chamfer_prep_kernel(
    const float* __restrict__ tpts, const float* __restrict__ spts,
    float* __restrict__ b4_t, float* __restrict__ n2_t,
    float* __restrict__ b4_s, float* __restrict__ n2_s, int total)
{
    int i = blockIdx.x * blockDim.x + threadIdx.x;
    if (i >= 2 * total) return;
    const float* pts;
    float *b4, *n2;
    int p;
    if (i < total) { pts = tpts; b4 = b4_t; n2 = n2_t; p = i; }
    else           { pts = spts; b4 = b4_s; n2 = n2_s; p = i - total; }
    float x = pts[p * 3 + 0];
    float y = pts[p * 3 + 1];
    float z = pts[p * 3 + 2];
    float q = x * x + y * y + z * z;
    b4[p * 4 + 0] = -2.0f * x;
    b4[p * 4 + 1] = -2.0f * y;
    b4[p * 4 + 2] = -2.0f * z;
    b4[p * 4 + 3] = q;
    n2[p] = q;
}

// ---------------------------------------------------------------------------
// One wave owns 16 rows of the A-cloud; sweeps all M columns of the B-cloud
// in 16-wide tiles via V_WMMA_F32_16X16X4_F32. Per outer iteration: 4 tiles,
// folded into the row-min accumulator with 16x v_min3_num_f32.
//
// A fragment (ISA 16x4 f32 layout): lane l -> row M = l&15;
//   lanes 0-15 hold K={0,1} = (x,y); lanes 16-31 hold K={2,3} = (z,1).
// B fragment (4x16): lane l -> col N = l&15;
//   lanes 0-15 hold K={0,1} = (-2sx,-2sy); lanes 16-31 K={2,3} = (-2sz,s2).
// C/D: VGPR i -> row i (lanes 0-15) / row i+8 (lanes 16-31), col = lane&15.
// ---------------------------------------------------------------------------
__global__ __launch_bounds__(256) void chamfer_min_kernel(
    const float* __restrict__ apts,   // A-cloud raw [BATCH, NPTS, 3]
    const float* __restrict__ an2,    // [BATCH*NPTS] squared norms of A-cloud
    const float* __restrict__ bb4,    // B-cloud packed [BATCH*NPTS*4]
    float* __restrict__ outmin)       // [BATCH*NPTS] sqrt(min d^2)
{
    const int b    = blockIdx.y;
    const int wave = threadIdx.x >> 5;
    const int lane = threadIdx.x & 31;
    const int half = lane >> 4;      // 0: K={0,1}, 1: K={2,3}
    const int l16  = lane & 15;
    const int rowBase = blockIdx.x * ROWS_PER_BLOCK + wave * ROWS_PER_WAVE;

    // Build the A fragment: both lane halves read the same template point.
    const int arow = b * NPTS + rowBase + l16;
    const float ax = apts[arow * 3 + 0];
    const float ay = apts[arow * 3 + 1];
    const float az = apts[arow * 3 + 2];
    v2f a;
    a.x = half ? az : ax;
    a.y = half ? 1.0f : ay;

    // Per-lane pointer into packed B-operand: column l16, K-pair (2*half).
    const float* bbase = bb4 + (size_t)b * NPTS * 4 + l16 * 4 + half * 2;

    float mn[8];
#pragma unroll
    for (int i = 0; i < 8; ++i) mn[i] = 3.0e38f;

    for (int col = 0; col < NPTS; col += 64) {   // 4 tiles per iteration
        __builtin_prefetch(bbase + (size_t)(col + 512) * 4, 0, 1);
        v2f b0 = *(const v2f*)(bbase + (size_t)(col +  0) * 4);
        v2f b1 = *(const v2f*)(bbase + (size_t)(col + 16) * 4);
        v2f b2 = *(const v2f*)(bbase + (size_t)(col + 32) * 4);
        v2f b3 = *(const v2f*)(bbase + (size_t)(col + 48) * 4);
        v8f z = {};
        v8f c0 = __builtin_amdgcn_wmma_f32_16x16x4_f32(false, a, false, b0,
                                                       (short)0, z, false, false);
        v8f c1 = __builtin_amdgcn_wmma_f32_16x16x4_f32(false, a, false, b1,
                                                       (short)0, z, false, false);
        v8f c2 = __builtin_amdgcn_wmma_f32_16x16x4_f32(false, a, false, b2,
                                                       (short)0, z, false, false);
        v8f c3 = __builtin_amdgcn_wmma_f32_16x16x4_f32(false, a, false, b3,
                                                       (short)0, z, false, false);
#pragma unroll
        for (int i = 0; i < 8; ++i) mn[i] = min3_asm(mn[i], c0[i], c1[i]);
#pragma unroll
        for (int i = 0; i < 8; ++i) mn[i] = min3_asm(mn[i], c2[i], c3[i]);
    }

    // Reduce across the 16 columns held by each lane-half. ds_swizzle XOR
    // masks 1,2,4,8 never cross the 16-lane boundary on wave32.
#pragma unroll
    for (int i = 0; i < 8; ++i) {
        SWZ_MIN(mn[i], SWZ_XOR1);
        SWZ_MIN(mn[i], SWZ_XOR2);
        SWZ_MIN(mn[i], SWZ_XOR4);
        SWZ_MIN(mn[i], SWZ_XOR8);
    }

    // Lane 0 holds rows rowBase..+7 (VGPR i -> row i); lane 16 rows +8..+15.
    if (l16 == 0) {
#pragma unroll
        for (int i = 0; i < 8; ++i) {
            int r = rowBase + half * 8 + i;
            float t2 = an2[b * NPTS + r];
            float d2 = fmaxf(mn[i] + t2, 0.0f);   // guard tiny negatives
            outmin[b * NPTS + r] = sqrtf(d2);
        }
    }
}

// ---------------------------------------------------------------------------
// Deterministic final reduction: fixed-order per-thread partial sums + LDS
// tree. out = 0.5 * (mean(sqrt_min_t) + mean(sqrt_min_s)).
// ---------------------------------------------------------------------------
__global__ __launch_bounds__(256) void chamfer_reduce_kernel(
    const float* __restrict__ mind_t, const float* __restrict__ mind_s,
    float* __restrict__ out, int total)
{
    __shared__ float s1[256];
    __shared__ float s2[256];
    int t = threadIdx.x;
    float acc1 = 0.0f, acc2 = 0.0f;
    for (int i = t; i < total; i += 256) {
        acc1 += mind_t[i];
        acc2 += mind_s[i];
    }
    s1[t] = acc1;
    s2[t] = acc2;
    __syncthreads();
    for (int off = 128; off > 0; off >>= 1) {
        if (t < off) {
            s1[t] += s1[t + off];
            s2[t] += s2[t + off];
        }
        __syncthreads();
    }
    if (t == 0) {
        float inv = 1.0f / (float)total;
        out[0] = 0.5f * (s1[0] * inv + s2[0] * inv);
    }
}

extern "C" void kernel_launch(void* const* d_in, const int* in_sizes, int n_in,
                              void* d_out, int out_size, void* d_ws, size_t ws_size,
                              hipStream_t stream) {
    const float* tmpl = (const float*)d_in[0];   // [4, 8192, 3]
    const float* src  = (const float*)d_in[1];   // [4, 8192, 3]
    float* out = (float*)d_out;

    const int total = BATCH * NPTS;              // 32768 points per cloud

    // Workspace carve-up (floats): 2*131072 + 2*32768 + 2*32768 = 393216 f
    float* ws     = (float*)d_ws;
    float* b4_t   = ws;                          // 131072
    float* b4_s   = b4_t + (size_t)total * 4;    // 131072
    float* n2_t   = b4_s + (size_t)total * 4;    // 32768
    float* n2_s   = n2_t + total;                // 32768
    float* mind_t = n2_s + total;                // 32768
    float* mind_s = mind_t + total;              // 32768

    chamfer_prep_kernel<<<(2 * total + 255) / 256, 256, 0, stream>>>(
        tmpl, src, b4_t, n2_t, b4_s, n2_s, total);

    dim3 grid(NPTS / ROWS_PER_BLOCK, BATCH);     // (64, 4)
    // Direction 0: for each template point, NN over source.
    chamfer_min_kernel<<<grid, 256, 0, stream>>>(tmpl, n2_t, b4_s, mind_t);
    // Direction 1: for each source point, NN over template.
    chamfer_min_kernel<<<grid, 256, 0, stream>>>(src, n2_s, b4_t, mind_s);

    chamfer_reduce_kernel<<<1, 256, 0, stream>>>(mind_t, mind_s, out, total);
}